// SemanticFilter_20658792694712
// MI455X (gfx1250) — compile-verified
//
#include <hip/hip_runtime.h>
#include <math.h>

typedef float v2f __attribute__((ext_vector_type(2)));
typedef float v4f __attribute__((ext_vector_type(4)));
typedef float v8f __attribute__((ext_vector_type(8)));

#define HDIM      768
#define NMAXR     64
#define BNUM      4096
#define QNUM      16384
#define SNUM      256
#define KC        64          // k-chunk (floats) staged in LDS
#define NCHUNK    12          // 768 / 64
#define ROWSTRIDE 68          // KC + 4 pad -> conflict-free ds_load_b64
#define KSTEPS    16          // KC / 4 wmma steps per chunk
#define WB_STEPS  192         // 768 / 4 total k-steps
#define INV_TEMP  (1.0f / 0.1f)

// ---------------------------------------------------------------------------
// Kernel 0: build the B-fragment stream for V_WMMA_F32_16X16X4_F32.
// B is 4x16 (KxN): col0 = query_w, col1 = wm_w[0:768], col2 = wm_w[768:1536],
// cols 3..15 = 0.  Per k-step, lane l holds W[k + 2*(l>>4) + {0,1}][l&15].
// Stream layout: wb[step*64 + lane*2 + {0,1}] -> coalesced b64 per wave.
// ---------------------------------------------------------------------------
__global__ __launch_bounds__(256) void wb_init_kernel(
    const float* __restrict__ qw, const float* __restrict__ wmw,
    float* __restrict__ wb) {
  int idx = blockIdx.x * 256 + threadIdx.x;
  if (idx >= WB_STEPS * 32) return;
  int step = idx >> 5, lane = idx & 31;
  int n = lane & 15, hi = lane >> 4;
  int kk = step * 4 + hi * 2;
  float x0 = 0.f, x1 = 0.f;
  if (n == 0)      { x0 = qw[kk];          x1 = qw[kk + 1]; }
  else if (n == 1) { x0 = wmw[kk];         x1 = wmw[kk + 1]; }
  else if (n == 2) { x0 = wmw[768 + kk];   x1 = wmw[768 + kk + 1]; }
  wb[step * 64 + lane * 2]     = x0;
  wb[step * 64 + lane * 2 + 1] = x1;
}

// ---------------------------------------------------------------------------
// Kernel 1: fused attention-pool + projection.  One block per b (4 waves,
// each owns a 16-row M-tile).  Streams the 64x768 tile through double-
// buffered LDS; each wave runs 192 V_WMMA_F32_16X16X4_F32 accumulating
// D[16,16] whose cols 0..2 are (score, dot wm_lo, dot wm_hi) per row.
// Then wave 0 does the masked softmax over 64 rows and emits a[b], t[b].
// ---------------------------------------------------------------------------
__global__ __launch_bounds__(128) void merge_proj_kernel(
    const float* __restrict__ padded, const int* __restrict__ lengths,
    const float* __restrict__ wb, float* __restrict__ Aout,
    float* __restrict__ Tout) {
  __shared__ float tile[2][NMAXR * ROWSTRIDE];
  __shared__ float dl[3][NMAXR];

  const int b    = blockIdx.x;
  const int tid  = threadIdx.x;
  const int lane = tid & 31;
  const int wave = tid >> 5;
  const size_t base = (size_t)b * NMAXR * HDIM;

  // staging pattern: 128 threads, 8 iterations of float4 (16 KB / chunk)
  const int lrow = tid >> 4;         // 0..7
  const int lcol = (tid & 15) * 4;   // 0..60

  v4f reg[8];
#pragma unroll
  for (int i = 0; i < 8; ++i)
    reg[i] = *(const v4f*)(padded + base + (size_t)(i * 8 + lrow) * HDIM + lcol);
#pragma unroll
  for (int i = 0; i < 8; ++i)
    *(v4f*)(&tile[0][(i * 8 + lrow) * ROWSTRIDE + lcol]) = reg[i];
  __syncthreads();

  v8f c = {0.f, 0.f, 0.f, 0.f, 0.f, 0.f, 0.f, 0.f};
  // A-fragment LDS address: row = 16*wave + (lane&15), k-subgroup = 2*(lane>>4)
  const int arow  = (16 * wave + (lane & 15)) * ROWSTRIDE + ((lane >> 4) * 2);
  const int boff  = lane * 2;

  for (int ch = 0; ch < NCHUNK; ++ch) {
    // prefetch next chunk into registers (overlaps WMMA below)
    if (ch + 1 < NCHUNK) {
#pragma unroll
      for (int i = 0; i < 8; ++i)
        reg[i] = *(const v4f*)(padded + base + (size_t)(i * 8 + lrow) * HDIM +
                               (ch + 1) * KC + lcol);
    }
    const float* tb   = tile[ch & 1];
    const float* wbch = wb + (size_t)(ch * KSTEPS) * 64;
#pragma unroll
    for (int ks = 0; ks < KSTEPS; ++ks) {
      v2f afrag = *(const v2f*)(tb + arow + ks * 4);
      v2f bfrag = *(const v2f*)(wbch + ks * 64 + boff);
      c = __builtin_amdgcn_wmma_f32_16x16x4_f32(
          /*neg_a=*/false, afrag, /*neg_b=*/false, bfrag,
          /*c_mod=*/(short)0, c, /*reuse_a=*/false, /*reuse_b=*/false);
    }
    if (ch + 1 < NCHUNK) {
      float* td = tile[(ch + 1) & 1];
#pragma unroll
      for (int i = 0; i < 8; ++i)
        *(v4f*)(&td[(i * 8 + lrow) * ROWSTRIDE + lcol]) = reg[i];
    }
    __syncthreads();
  }

  // D layout: lane n = lane&15 holds column n; VGPR v holds row v + 8*(lane>>4)
  {
    int n = lane & 15, hi = lane >> 4;
    if (n < 3) {
#pragma unroll
      for (int v = 0; v < 8; ++v)
        dl[n][16 * wave + v + 8 * hi] = c[v];
    }
  }
  __syncthreads();

  if (wave == 0) {
    const int len = lengths[b];
    const int r0 = lane, r1 = lane + 32;
    float s0 = (r0 < len) ? dl[0][r0] : -__builtin_inff();
    float s1 = (r1 < len) ? dl[0][r1] : -__builtin_inff();
    float m = fmaxf(s0, s1);
#pragma unroll
    for (int off = 16; off >= 1; off >>= 1) m = fmaxf(m, __shfl_xor(m, off, 32));
    float e0 = (r0 < len) ? expf(s0 - m) : 0.f;
    float e1 = (r1 < len) ? expf(s1 - m) : 0.f;
    float se = e0 + e1;
    float av = e0 * dl[1][r0] + e1 * dl[1][r1];
    float tv = e0 * dl[2][r0] + e1 * dl[2][r1];
#pragma unroll
    for (int off = 16; off >= 1; off >>= 1) {
      se += __shfl_xor(se, off, 32);
      av += __shfl_xor(av, off, 32);
      tv += __shfl_xor(tv, off, 32);
    }
    if (lane == 0) {
      float inv = 1.0f / se;
      Aout[b] = av * inv;
      Tout[b] = tv * inv;
    }
  }
}

// ---------------------------------------------------------------------------
// Kernel 2: logits[q] = (a[cidx[type[q]]] + t[pidx[q]] + wm_b) / TEMP
// ---------------------------------------------------------------------------
__global__ __launch_bounds__(256) void logits_kernel(
    const float* __restrict__ Aarr, const float* __restrict__ Tarr,
    const int* __restrict__ pidx, const int* __restrict__ cidx,
    const int* __restrict__ types, const float* __restrict__ wmb,
    float* __restrict__ logits) {
  int q = blockIdx.x * 256 + threadIdx.x;
  if (q >= QNUM) return;
  int t = types[q];
  float s = Aarr[cidx[t]] + Tarr[pidx[q]] + wmb[0];
  logits[q] = s * INV_TEMP;
}

// ---------------------------------------------------------------------------
// Kernel 3: per-segment LSE terms.  One block per segment; deterministic
// fixed-tree reductions (no float atomics -> graph replay stable).
// ---------------------------------------------------------------------------
__global__ __launch_bounds__(256) void segment_kernel(
    const float* __restrict__ logits, const int* __restrict__ types,
    const int* __restrict__ labels, float* __restrict__ contrib,
    float* __restrict__ haspos) {
  __shared__ float ra[256], rb[256], rc[256];
  const int s = blockIdx.x, tid = threadIdx.x;

  float gmax = -__builtin_inff(), pmax = -__builtin_inff(), pcnt = 0.f;
  for (int q = tid; q < QNUM; q += 256) {
    if (types[q] == s) {
      float l = logits[q];
      gmax = fmaxf(gmax, l);
      if (labels[q] == 1) { pmax = fmaxf(pmax, l); pcnt += 1.f; }
    }
  }
  ra[tid] = gmax; rb[tid] = pmax; rc[tid] = pcnt;
  __syncthreads();
  for (int off = 128; off >= 1; off >>= 1) {
    if (tid < off) {
      ra[tid] = fmaxf(ra[tid], ra[tid + off]);
      rb[tid] = fmaxf(rb[tid], rb[tid + off]);
      rc[tid] += rc[tid + off];
    }
    __syncthreads();
  }
  const float gm = ra[0], pm = rb[0], pc = rc[0];
  const float gs = (gm > -__builtin_inff()) ? gm : 0.f;
  const float ps = (pm > -__builtin_inff()) ? pm : 0.f;
  __syncthreads();

  float sa = 0.f, sp = 0.f;
  for (int q = tid; q < QNUM; q += 256) {
    if (types[q] == s) {
      float l = logits[q];
      sa += expf(l - gs);
      if (labels[q] == 1) sp += expf(l - ps);
    }
  }
  ra[tid] = sa; rb[tid] = sp;
  __syncthreads();
  for (int off = 128; off >= 1; off >>= 1) {
    if (tid < off) { ra[tid] += ra[tid + off]; rb[tid] += rb[tid + off]; }
    __syncthreads();
  }
  if (tid == 0) {
    int has = (pc > 0.f);
    float lse_all = logf(has ? ra[0] : 1.f) + gs;
    float lse_pos = logf(has ? rb[0] : 1.f) + ps;
    contrib[s] = has ? (lse_all - lse_pos) : 0.f;
    haspos[s]  = has ? 1.f : 0.f;
  }
}

// ---------------------------------------------------------------------------
// Kernel 4: final scalar loss.
// ---------------------------------------------------------------------------
__global__ __launch_bounds__(256) void final_kernel(
    const float* __restrict__ contrib, const float* __restrict__ haspos,
    float* __restrict__ out) {
  __shared__ float ra[256], rb[256];
  int tid = threadIdx.x;
  ra[tid] = contrib[tid];
  rb[tid] = haspos[tid];
  __syncthreads();
  for (int off = 128; off >= 1; off >>= 1) {
    if (tid < off) { ra[tid] += ra[tid + off]; rb[tid] += rb[tid + off]; }
    __syncthreads();
  }
  if (tid == 0) out[0] = ra[0] / fmaxf(rb[0], 1.f);
}

// ---------------------------------------------------------------------------
extern "C" void kernel_launch(void* const* d_in, const int* in_sizes, int n_in,
                              void* d_out, int out_size, void* d_ws,
                              size_t ws_size, hipStream_t stream) {
  const float* padded = (const float*)d_in[0];  // [B, 64, 768] f32
  const float* qw     = (const float*)d_in[1];  // [768]
  // d_in[2] = query_b: constant per-row shift, softmax-invariant -> unused
  const float* wmw    = (const float*)d_in[3];  // [1536]
  const float* wmb    = (const float*)d_in[4];  // [1]
  const int*   lens   = (const int*)d_in[5];    // [B]
  const int*   pidx   = (const int*)d_in[6];    // [Q]
  const int*   cidx   = (const int*)d_in[7];    // [S]
  const int*   types  = (const int*)d_in[8];    // [Q]
  const int*   labels = (const int*)d_in[9];    // [Q,1]
  float*       out    = (float*)d_out;

  float* ws      = (float*)d_ws;
  float* wb      = ws;                     // 192*64 = 12288 floats
  float* Aarr    = wb + WB_STEPS * 64;     // 4096
  float* Tarr    = Aarr + BNUM;            // 4096
  float* logits  = Tarr + BNUM;            // 16384
  float* contrib = logits + QNUM;          // 256
  float* haspos  = contrib + SNUM;         // 256

  wb_init_kernel<<<(WB_STEPS * 32 + 255) / 256, 256, 0, stream>>>(qw, wmw, wb);
  merge_proj_kernel<<<BNUM, 128, 0, stream>>>(padded, lens, wb, Aarr, Tarr);
  logits_kernel<<<QNUM / 256, 256, 0, stream>>>(Aarr, Tarr, pidx, cidx, types,
                                                wmb, logits);
  segment_kernel<<<SNUM, 256, 0, stream>>>(logits, types, labels, contrib,
                                           haspos);
  final_kernel<<<1, 256, 0, stream>>>(contrib, haspos, out);
}